// LeftPool_1417339208203
// MI455X (gfx1250) — compile-verified
//
#include <hip/hip_runtime.h>

// LeftPool: out[..., w] = max(x[..., w:]) — reverse (suffix) cummax along the
// last axis (width = 128) of a (16, 256, 128, 128) f32 tensor.
//
// Strategy (MI455X, wave32):
//   - One wave per row of 128 floats: lane l holds 4 contiguous elems (b128).
//   - In-register reverse cummax of the 4 local elements.
//   - Wave-wide suffix-max scan via __shfl_down (5 steps, wave32).
//   - Exclusive suffix (lanes strictly to the right) via one more shfl_down.
//   - Combine, store b128.
//   - Non-temporal load/store: both 256 MB streams are single-touch and
//     exceed the 192 MB L2, so TH=NT hints keep L2 clean.
//
// Roofline: 512 MB total traffic / 23.3 TB/s ≈ 22 us floor; compute is
// negligible (~15 VALU + 6 DS-permute ops per 512 B wave transaction).

typedef float v4f __attribute__((ext_vector_type(4)));

#define WAVES_PER_BLOCK 8
#define BLOCK_THREADS   (WAVES_PER_BLOCK * 32)

__global__ __launch_bounds__(BLOCK_THREADS) void
leftpool_rowscan_kernel(const float* __restrict__ x,
                        float* __restrict__ out,
                        int nrows)
{
    const int lane = threadIdx.x & 31;
    const int wave = threadIdx.x >> 5;
    const long long row = (long long)blockIdx.x * WAVES_PER_BLOCK + wave;
    if (row >= nrows) return;  // uniform per wave -> whole wave exits together

    const v4f* __restrict__ src =
        reinterpret_cast<const v4f*>(x + row * 128) + lane;
    v4f* __restrict__ dst =
        reinterpret_cast<v4f*>(out + row * 128) + lane;

    // Single-touch streaming load (global_load_b128, non-temporal).
    v4f v = __builtin_nontemporal_load(src);

    // Reverse cummax within the lane's 4 contiguous elements.
    const float y3 = v.w;
    const float y2 = fmaxf(v.z, y3);
    const float y1 = fmaxf(v.y, y2);
    const float y0 = fmaxf(v.x, y1);  // = max of this lane's whole segment

    // Inclusive suffix-max across the wave (lane l gets max over lanes >= l).
    float s = y0;
#pragma unroll
    for (int off = 1; off < 32; off <<= 1) {
        float o = __shfl_down(s, off, 32);   // wave32-width shuffle
        s = (lane + off < 32) ? fmaxf(s, o) : s;
    }

    // Exclusive suffix: max over lanes strictly greater than this one.
    float e = __shfl_down(s, 1, 32);
    e = (lane == 31) ? -__builtin_huge_valf() : e;

    v4f r;
    r.x = fmaxf(y0, e);
    r.y = fmaxf(y1, e);
    r.z = fmaxf(y2, e);
    r.w = fmaxf(y3, e);

    // Single-touch streaming store (global_store_b128, non-temporal).
    __builtin_nontemporal_store(r, dst);
}

extern "C" void kernel_launch(void* const* d_in, const int* in_sizes, int n_in,
                              void* d_out, int out_size, void* d_ws, size_t ws_size,
                              hipStream_t stream) {
    (void)in_sizes; (void)n_in; (void)d_ws; (void)ws_size;

    const float* x  = (const float*)d_in[0];
    float*      out = (float*)d_out;

    // out_size = 16*256*128*128 = 67,108,864 elements; width is fixed at 128.
    const int nrows  = out_size / 128;                       // 524,288 rows
    const int blocks = (nrows + WAVES_PER_BLOCK - 1) / WAVES_PER_BLOCK;

    leftpool_rowscan_kernel<<<blocks, BLOCK_THREADS, 0, stream>>>(x, out, nrows);
}